// HNet3_74801150427700
// MI455X (gfx1250) — compile-verified
//
#include <hip/hip_runtime.h>
#include <stdint.h>

// HNet3 masked softmax over groups of 8:
//   mid = sorted(x)[3] (ascending)  -> element of descending rank 4
//   out = softmax(x) * (x > mid)
//
// Memory-bound: 512 MB total traffic -> ~22 us floor @ 23.3 TB/s.
// Data path: gfx1250 async global->LDS B128 copies (non-temporal, streaming),
// double-buffered per wave and fenced with s_wait_asynccnt; output written with
// non-temporal b128 stores so the two 256 MB streams don't thrash the 192 MB L2.

#define NPG 8               // elements per group (row)
#define TPB 256             // threads per block (8 waves of 32)
#define ROWS_PER_BLOCK (2 * TPB)

// Native clang vector type: required by __builtin_nontemporal_store and gives
// b128 ds/global accesses.
typedef float vfloat4 __attribute__((ext_vector_type(4)));

__device__ __forceinline__ void cexch(float& a, float& b) {
    float lo = fminf(a, b);
    float hi = fmaxf(a, b);
    a = lo;
    b = hi;
}

__device__ __forceinline__ void process_row(const float* __restrict__ lrow,
                                            float* __restrict__ dst,
                                            bool valid) {
    // Two ds_load_b128 from this lane's private LDS slot.
    vfloat4 a = *reinterpret_cast<const vfloat4*>(lrow);
    vfloat4 b = *reinterpret_cast<const vfloat4*>(lrow + 4);
    float x0 = a.x, x1 = a.y, x2 = a.z, x3 = a.w;
    float x4 = b.x, x5 = b.y, x6 = b.z, x7 = b.w;

    // Batcher odd-even mergesort network for 8 (19 compare-exchanges).
    float s0 = x0, s1 = x1, s2 = x2, s3 = x3, s4 = x4, s5 = x5, s6 = x6, s7 = x7;
    cexch(s0, s1); cexch(s2, s3); cexch(s4, s5); cexch(s6, s7);
    cexch(s0, s2); cexch(s1, s3); cexch(s4, s6); cexch(s5, s7);
    cexch(s1, s2); cexch(s5, s6);
    cexch(s0, s4); cexch(s1, s5); cexch(s2, s6); cexch(s3, s7);
    cexch(s2, s4); cexch(s3, s5);
    cexch(s1, s2); cexch(s3, s4); cexch(s5, s6);

    const float mid = s3;  // ascending index 3 == descending rank num_per_group//2
    const float mx  = s7;  // row max (free from the network) for stable softmax

    float e0 = __expf(x0 - mx), e1 = __expf(x1 - mx);
    float e2 = __expf(x2 - mx), e3 = __expf(x3 - mx);
    float e4 = __expf(x4 - mx), e5 = __expf(x5 - mx);
    float e6 = __expf(x6 - mx), e7 = __expf(x7 - mx);
    float sum = ((e0 + e1) + (e2 + e3)) + ((e4 + e5) + (e6 + e7));
    float r = 1.0f / sum;

    vfloat4 o0, o1;
    o0.x = (x0 > mid) ? e0 * r : 0.0f;
    o0.y = (x1 > mid) ? e1 * r : 0.0f;
    o0.z = (x2 > mid) ? e2 * r : 0.0f;
    o0.w = (x3 > mid) ? e3 * r : 0.0f;
    o1.x = (x4 > mid) ? e4 * r : 0.0f;
    o1.y = (x5 > mid) ? e5 * r : 0.0f;
    o1.z = (x6 > mid) ? e6 * r : 0.0f;
    o1.w = (x7 > mid) ? e7 * r : 0.0f;

    if (valid) {
        // Non-temporal b128 stores: streaming output, never re-read.
        __builtin_nontemporal_store(o0, reinterpret_cast<vfloat4*>(dst));
        __builtin_nontemporal_store(o1, reinterpret_cast<vfloat4*>(dst + 4));
    }
}

__global__ __launch_bounds__(TPB) void HNet3_masked_softmax_kernel(
    const float* __restrict__ in, float* __restrict__ out, int rows) {
    __shared__ __align__(16) float smem[2][TPB * NPG];  // 2 x 8 KB tiles

    const int tid = threadIdx.x;
    const long long rbase = (long long)blockIdx.x * ROWS_PER_BLOCK;
    const long long r0 = rbase + tid;
    const long long r1 = rbase + TPB + tid;
    // Clamp load addresses (stores are predicated) for a safe partial tail.
    const long long c0 = (r0 < rows) ? r0 : (long long)rows - 1;
    const long long c1 = (r1 < rows) ? r1 : (long long)rows - 1;

    const unsigned int l0 = (unsigned int)(uintptr_t)&smem[0][tid * NPG];
    const unsigned int l1 = (unsigned int)(uintptr_t)&smem[1][tid * NPG];
    const unsigned long long g0 = (unsigned long long)(uintptr_t)(in + c0 * NPG);
    const unsigned long long g1 = (unsigned long long)(uintptr_t)(in + c1 * NPG);

    // Async DMA global->LDS, 16 B per lane per op; the instruction offset applies
    // to BOTH the LDS and global address, so offset:16 moves the row's second
    // half. TH_LOAD_NT: streaming input, don't retain in caches.
    asm volatile("global_load_async_to_lds_b128 %0, %1, off th:TH_LOAD_NT"
                 :: "v"(l0), "v"(g0) : "memory");
    asm volatile("global_load_async_to_lds_b128 %0, %1, off offset:16 th:TH_LOAD_NT"
                 :: "v"(l0), "v"(g0) : "memory");
    asm volatile("global_load_async_to_lds_b128 %0, %1, off th:TH_LOAD_NT"
                 :: "v"(l1), "v"(g1) : "memory");
    asm volatile("global_load_async_to_lds_b128 %0, %1, off offset:16 th:TH_LOAD_NT"
                 :: "v"(l1), "v"(g1) : "memory");

    // Tile 0 ready (async ops complete in order); tile 1 still in flight.
    asm volatile("s_wait_asynccnt 0x2" ::: "memory");
    process_row(&smem[0][tid * NPG], out + r0 * NPG, r0 < (long long)rows);

    // Drain remaining async copies, then finish tile 1.
    asm volatile("s_wait_asynccnt 0x0" ::: "memory");
    process_row(&smem[1][tid * NPG], out + r1 * NPG, r1 < (long long)rows);
}

extern "C" void kernel_launch(void* const* d_in, const int* in_sizes, int n_in,
                              void* d_out, int out_size, void* d_ws, size_t ws_size,
                              hipStream_t stream) {
    (void)n_in; (void)d_ws; (void)ws_size; (void)out_size;
    const float* in = (const float*)d_in[0];
    float* out = (float*)d_out;
    // d_in[1] is num_per_group (== 8); the kernel is specialized for 8.
    const long long total = (long long)in_sizes[0];
    const int rows = (int)(total / NPG);   // 8,388,608
    const int blocks = (rows + ROWS_PER_BLOCK - 1) / ROWS_PER_BLOCK;  // 16,384
    HNet3_masked_softmax_kernel<<<dim3(blocks), dim3(TPB), 0, stream>>>(in, out, rows);
}